// HGCN_44409961841006
// MI455X (gfx1250) — compile-verified
//
#include <hip/hip_runtime.h>

// ---------------------------------------------------------------------------
// Sizes fixed by the reference
// ---------------------------------------------------------------------------
#define B_DLG   200
#define L_UTT   50
#define NMODAL  3
#define NDIM    128
#define NHID    128
#define NUMK    4
#define NSPK    2
#define NNODES  (B_DLG * NMODAL * L_UTT)     // 30000
#define OUTSTR  (NMODAL * 2 * NDIM)          // 768 floats per output row

typedef __attribute__((ext_vector_type(2))) float v2f;
typedef __attribute__((ext_vector_type(8))) float v8f;

#ifndef __has_builtin
#define __has_builtin(x) 0
#endif

// gfx1250 async global->LDS copy path (ASYNCcnt), guarded for toolchain drift
#if defined(__gfx1250__) && __has_builtin(__builtin_amdgcn_global_load_async_to_lds_b128)
#define USE_ASYNC_LDS 1
#else
#define USE_ASYNC_LDS 0
#endif

#if USE_ASYNC_LDS
// Builtin signature (from hipcc diagnostic): param0 = v4i in AS1 ("__device__"),
// param1 = v4i in AS3, both non-const; then two constant ints (offset, cpol).
typedef int v4i_vs __attribute__((vector_size(4 * sizeof(int))));
typedef __attribute__((address_space(1))) v4i_vs* as1_v4i_ptr;
typedef __attribute__((address_space(3))) v4i_vs* as3_v4i_ptr;
#endif

// ---------------------------------------------------------------------------
// Kernel 1: build feats in node order [l_b(50); a_b(50); v_b(50)] per dialogue,
// adding speaker embedding to the 'l' block, and simultaneously write the
// feats half of the final output (use_residue + reverse_features layout).
// One float4 per thread, fully coalesced.
// ---------------------------------------------------------------------------
__global__ __launch_bounds__(256) void build_feats_kernel(
    const float* __restrict__ a, const float* __restrict__ v,
    const float* __restrict__ l, const float* __restrict__ qmask,
    const float* __restrict__ spk_emb,
    float* __restrict__ feats, float* __restrict__ out)
{
  int g = blockIdx.x * 256 + threadIdx.x;         // one float4 each
  if (g >= NNODES * (NDIM / 4)) return;
  int row = g >> 5;                               // node index 0..29999
  int c   = (g & 31) << 2;                        // dim offset 0,4,...,124
  int b   = row / (NMODAL * L_UTT);
  int rem = row - b * (NMODAL * L_UTT);
  int m   = rem / L_UTT;
  int t   = rem - m * L_UTT;
  int u   = b * L_UTT + t;                        // utterance index

  const float* src = (m == 0) ? l : ((m == 1) ? a : v);
  float4 val = *(const float4*)(src + (size_t)u * NDIM + c);

  if (m == 0) {
    // argmax over 2 speakers; strict > matches jnp.argmax first-max tiebreak
    float q0 = qmask[((size_t)t * B_DLG + b) * NSPK + 0];
    float q1 = qmask[((size_t)t * B_DLG + b) * NSPK + 1];
    int spk = (q1 > q0) ? 1 : 0;
    float4 e = *(const float4*)(spk_emb + spk * NDIM + c);
    val.x += e.x; val.y += e.y; val.z += e.z; val.w += e.w;
  }

  *(float4*)(feats + (size_t)row * NDIM + c) = val;
  // output: out[(b*50+t)*768 + m*256 + d]  (feats half of residue concat)
  *(float4*)(out + (size_t)u * OUTSTR + m * (2 * NDIM) + c) = val;
}

// ---------------------------------------------------------------------------
// Kernel 2: x = feats @ W1^T + b1 via V_WMMA_F32_16X16X4_F32.
// Each wave owns a 16-row strip and all 128 output columns (8 accumulators),
// reusing the A fragment across the 8 N-tiles each K step.
//
// f32 A layout (16x4): lanes 0-15 hold (M=lane, K=4k+{0,1}), lanes 16-31
// hold (M=lane-16, K=4k+{2,3}) -> one aligned float2 load per step.
// B (4x16) mirrors with N in place of M; B[k][n] = W1[n][k], so B fragments
// are also contiguous float2 loads from W1 rows.
// ---------------------------------------------------------------------------
__global__ __launch_bounds__(256) void gemm_wmma_kernel(
    const float* __restrict__ feats, const float* __restrict__ W1,
    const float* __restrict__ b1, float* __restrict__ x)
{
  const int wave = threadIdx.x >> 5;
  const int lane = threadIdx.x & 31;
  const int mt   = blockIdx.x * 8 + wave;
  if (mt >= NNODES / 16) return;                  // wave-uniform exit, EXEC stays full

  const int hi   = lane >> 4;                     // half-wave select
  const int lr   = lane & 15;
  const int row0 = mt * 16;

  const float* arow = feats + (size_t)(row0 + lr) * NDIM + 2 * hi;

  v8f acc[8] = {};

  for (int kk = 0; kk < 32; ++kk) {               // K = 128 = 32 * 4
    v2f av = *(const v2f*)(arow + kk * 4);
#pragma unroll
    for (int nt = 0; nt < 8; ++nt) {
      v2f bv = *(const v2f*)(W1 + (size_t)(nt * 16 + lr) * NDIM + kk * 4 + 2 * hi);
      acc[nt] = __builtin_amdgcn_wmma_f32_16x16x4_f32(
          /*neg_a=*/false, av, /*neg_b=*/false, bv,
          /*c_mod=*/(short)0, acc[nt], /*reuse_a=*/false, /*reuse_b=*/false);
    }
  }

  // C/D layout: VGPR j holds (M = row0 + j + 8*hi, N = nt*16 + lr)
#pragma unroll
  for (int nt = 0; nt < 8; ++nt) {
    float bias = b1[nt * 16 + lr];
#pragma unroll
    for (int j = 0; j < 8; ++j) {
      int r = row0 + j + 8 * hi;
      x[(size_t)r * NHID + nt * 16 + lr] = acc[nt][j] + bias;
    }
  }
}

// ---------------------------------------------------------------------------
// Kernel 3: 4 hypergraph-conv iterations, LDS-resident per dialogue.
// Every node has deg == 51 (49 within-modality + 2 cross-modal), so:
//   agg[m,t] = (S[m] + T[t] - 2*x[m,t]) / 51
// with S[m] = sum_t x[m,t,:], T[t] = sum_m x[m,t,:].
// One 256-thread block per dialogue; 150x128 floats (76.8 KB) of x live in
// LDS across all 4 iterations (CDNA5: 320 KB LDS per workgroup).
// Staging uses gfx1250 async global->LDS copies (ASYNCcnt) when available.
// Finally writes the x half of the output.
// ---------------------------------------------------------------------------
__global__ __launch_bounds__(256) void hgcn_iter_kernel(
    const float* __restrict__ xin, const float* __restrict__ kappas,
    float* __restrict__ out)
{
  __shared__ float xs[NMODAL * L_UTT * NDIM];     // 19200 floats = 76.8 KB
  __shared__ float S[NMODAL * NDIM];              // 384 floats

  const int b   = blockIdx.x;
  const int tid = threadIdx.x;
  const float* xg = xin + (size_t)b * NMODAL * L_UTT * NDIM;

#if USE_ASYNC_LDS
  {
    // 19200 floats = 4800 float4 chunks: 18 full rounds of 256 + 192 extra.
    // Flat LDS address low 32 bits == LDS byte offset (ISA §10.2 aperture map).
    const float4* g4 = (const float4*)xg;
    for (int j = 0; j < 18; ++j) {
      int idx = tid + j * 256;
      __builtin_amdgcn_global_load_async_to_lds_b128(
          (as1_v4i_ptr)(uintptr_t)(g4 + idx),
          (as3_v4i_ptr)(uint32_t)(uintptr_t)(xs + idx * 4),
          0, 0);
    }
    if (tid < 192) {
      int idx = tid + 18 * 256;
      __builtin_amdgcn_global_load_async_to_lds_b128(
          (as1_v4i_ptr)(uintptr_t)(g4 + idx),
          (as3_v4i_ptr)(uint32_t)(uintptr_t)(xs + idx * 4),
          0, 0);
    }
    asm volatile("s_wait_asynccnt 0" ::: "memory");
  }
#else
#pragma unroll
  for (int j = 0; j < 75; ++j)                    // 19200 / 256 = 75 per thread
    xs[tid + j * 256] = xg[tid + j * 256];
#endif
  __syncthreads();

  const float inv_deg = 1.0f / 51.0f;

  for (int k = 0; k < NUMK; ++k) {
    float kappa = kappas[k];

    // per-(modality, dim) row sums over the 50 utterances (384 sums, 256 thr)
    for (int idx = tid; idx < NMODAL * NDIM; idx += 256) {
      int m = idx >> 7, d = idx & 127;
      float s = 0.0f;
#pragma unroll
      for (int t = 0; t < L_UTT; ++t)
        s += xs[(m * L_UTT + t) * NDIM + d];
      S[idx] = s;
    }
    __syncthreads();

    float rnew[75];
#pragma unroll
    for (int j = 0; j < 75; ++j) {
      int idx = tid + j * 256;
      int n = idx >> 7, d = idx & 127;
      int m = n / L_UTT, t = n - m * L_UTT;
      float xv = xs[idx];
      float Tv = xs[t * NDIM + d]
               + xs[(L_UTT + t) * NDIM + d]
               + xs[(2 * L_UTT + t) * NDIM + d];
      float agg = (S[m * NDIM + d] + Tv - 2.0f * xv) * inv_deg;
      float xn  = xv + kappa * agg;
      rnew[j] = xn > 0.0f ? xn : 0.0f;
    }
    __syncthreads();
#pragma unroll
    for (int j = 0; j < 75; ++j)
      xs[tid + j * 256] = rnew[j];
    __syncthreads();
  }

  // write x half of output: out[(b*50+t)*768 + m*256 + 128 + d]
#pragma unroll
  for (int j = 0; j < 75; ++j) {
    int idx = tid + j * 256;
    int n = idx >> 7, d = idx & 127;
    int m = n / L_UTT, t = n - m * L_UTT;
    out[((size_t)(b * L_UTT + t)) * OUTSTR + m * (2 * NDIM) + NDIM + d] = xs[idx];
  }
}

// ---------------------------------------------------------------------------
// Host entry
// Inputs: 0=a 1=v 2=l 3=qmask 4=W1 5=b1 6=speaker_emb 7=kappas 8=edge_index
//         9=epoch  (edge_index/epoch unused: graph structure is closed-form)
// ws layout: feats [30000*128] f32, x [30000*128] f32  (30.72 MB total)
// ---------------------------------------------------------------------------
extern "C" void kernel_launch(void* const* d_in, const int* in_sizes, int n_in,
                              void* d_out, int out_size, void* d_ws, size_t ws_size,
                              hipStream_t stream) {
  const float* a        = (const float*)d_in[0];
  const float* v        = (const float*)d_in[1];
  const float* l        = (const float*)d_in[2];
  const float* qmask    = (const float*)d_in[3];
  const float* W1       = (const float*)d_in[4];
  const float* b1       = (const float*)d_in[5];
  const float* spk_emb  = (const float*)d_in[6];
  const float* kappas   = (const float*)d_in[7];
  float* out   = (float*)d_out;
  float* feats = (float*)d_ws;
  float* xbuf  = feats + (size_t)NNODES * NDIM;

  int n4 = NNODES * (NDIM / 4);                        // 960000 float4 threads
  build_feats_kernel<<<(n4 + 255) / 256, 256, 0, stream>>>(
      a, v, l, qmask, spk_emb, feats, out);

  int mtiles = NNODES / 16;                            // 1875
  gemm_wmma_kernel<<<(mtiles + 7) / 8, 256, 0, stream>>>(feats, W1, b1, xbuf);

  hgcn_iter_kernel<<<B_DLG, 256, 0, stream>>>(xbuf, kappas, out);
}